// DeltaModulator_42923903156351
// MI455X (gfx1250) — compile-verified
//
#include <hip/hip_runtime.h>

typedef __attribute__((ext_vector_type(4))) float v4f;

namespace {
constexpr int C_IN    = 252;                      // input row length
constexpr int NSTEPS  = 232;                      // C_IN - SCAN_LEN_OFFSET(20)
constexpr int C_OUT   = 2 * NSTEPS + (C_IN - NSTEPS); // 484
constexpr int RPB     = 256;                      // rows per block == threads per block (8 wave32)
constexpr int TCH     = 32;                       // time chunk staged in LDS
constexpr int LDS_STR = TCH + 1;                  // 33 (odd) -> conflict-free per-row reads
constexpr float INITIAL_DELTA = 0.10f;
constexpr float DELTA_LOW     = 0.02f;
constexpr float DELTA_HIGH    = 0.10f;
}

#if __has_builtin(__builtin_amdgcn_global_load_async_to_lds_b32)
#define USE_ASYNC_LDS 1
#else
#define USE_ASYNC_LDS 0
#endif

typedef __attribute__((address_space(1))) int g_int;   // global
typedef __attribute__((address_space(3))) int l_int;   // LDS

__device__ __forceinline__ void wait_asynccnt0() {
#if __has_builtin(__builtin_amdgcn_s_wait_asynccnt)
    __builtin_amdgcn_s_wait_asynccnt(0);
#else
    asm volatile("s_wait_asynccnt 0x0" ::: "memory");
#endif
}

__global__ __launch_bounds__(RPB)
void delta_mod_kernel(const float* __restrict__ x, float* __restrict__ out, int nrows)
{
    __shared__ float tile[RPB * LDS_STR];         // 256*33*4 = 33,792 B of 320 KB/WGP

    const int tid     = threadIdx.x;
    const int rowBase = blockIdx.x * RPB;
    const int row     = rowBase + tid;
    const int lane    = tid & 31;                 // wave32 lane
    const int wv      = tid >> 5;                 // wave id 0..7

    // scan state (registers)
    float dc = 0.0f, delta = INITIAL_DELTA, tc = 0.0f, ntc = 0.0f;

    float*       outRow = out + (long long)row * C_OUT;
    const float* myRow  = &tile[tid * LDS_STR];

    for (int t0 = 0; t0 < C_IN; t0 += TCH) {
        const int tw = (C_IN - t0 < TCH) ? (C_IN - t0) : TCH;  // 32 x7, then 28

        __syncthreads();  // previous chunk's LDS reads are done before overwrite

        // Cooperative staged copy: each wave covers one row-chunk per issue,
        // lanes span 32 consecutive floats => one 128B coalesced request.
        for (int j = 0; j < RPB / 8; ++j) {
            const int r  = wv + 8 * j;
            const int gr = rowBase + r;
            if (lane < tw && gr < nrows) {
                const float* src = x + (long long)gr * C_IN + t0 + lane;
#if USE_ASYNC_LDS
                // CDNA5 direct global->LDS async DMA (no VGPR bounce), ASYNCcnt-tracked
                __builtin_amdgcn_global_load_async_to_lds_b32(
                    (g_int*)src,
                    (l_int*)(tile + r * LDS_STR + lane),
                    0, 0);
#else
                tile[r * LDS_STR + lane] = *src;
#endif
                if (t0 + TCH < C_IN)
                    __builtin_prefetch(src + TCH, 0, 1);   // global_prefetch_b8 (next chunk)
            }
        }
#if USE_ASYNC_LDS
        wait_asynccnt0();   // all async global->LDS transfers of this wave landed
#endif
        __syncthreads();

        if (row >= nrows) continue;

        // Scan this chunk; 4 steps unrolled -> 16B packed non-temporal stores.
        for (int t = 0; t < tw; t += 4) {
            const int gt = t0 + t;
            if (gt < NSTEPS) {
                v4f up4, dn4;
                #pragma unroll
                for (int k = 0; k < 4; ++k) {
                    const float xi   = myRow[t + k];
                    const bool up    = xi > dc + delta;
                    const bool dn    = xi < dc - delta;
                    const bool trig  = up || dn;
                    tc    = trig ? tc + 1.0f : 0.0f;
                    ntc   = trig ? 0.0f : ntc + 1.0f;
                    delta = (tc  >= 3.0f) ? DELTA_LOW  : delta;
                    delta = (ntc >= 3.0f) ? DELTA_HIGH : delta;
                    dc    = trig ? xi : dc;
                    up4[k] = up ? 1.0f : 0.0f;
                    dn4[k] = dn ? 1.0f : 0.0f;
                }
                // UP at [gt], DN at [NSTEPS+gt]; both 16B aligned (484*4 = 121*16).
                __builtin_nontemporal_store(up4, (v4f*)(outRow + gt));
                __builtin_nontemporal_store(dn4, (v4f*)(outRow + NSTEPS + gt));
            } else {
                // extras: gt in [232,252) -> out offset 464+(gt-232) == NSTEPS+gt
                v4f e;
                #pragma unroll
                for (int k = 0; k < 4; ++k) e[k] = myRow[t + k];
                __builtin_nontemporal_store(e, (v4f*)(outRow + NSTEPS + gt));
            }
        }
    }
}

extern "C" void kernel_launch(void* const* d_in, const int* in_sizes, int n_in,
                              void* d_out, int out_size, void* d_ws, size_t ws_size,
                              hipStream_t stream)
{
    const float* x   = (const float*)d_in[0];
    float*       out = (float*)d_out;
    const int nrows  = in_sizes[0] / C_IN;        // 128*1024 = 131072
    const int blocks = (nrows + RPB - 1) / RPB;   // 512
    delta_mod_kernel<<<blocks, RPB, 0, stream>>>(x, out, nrows);
}